// VQEncoder_90486370992355
// MI455X (gfx1250) — compile-verified
//
#include <hip/hip_runtime.h>
#include <hip/hip_bf16.h>

// CDNA5 / gfx1250 VQ-VAE encoder.
// fp32 WMMA (V_WMMA_F32_16X16X4_F32) for all GEMM work: 4 convs as implicit
// GEMM (M = Cout, N = B*OH*OW, K = Cin*16) and the VQ score GEMM
// (M = code, N = zf-row, K = 256).  argmin||z-c||^2 == argmax(z.c - 0.5||c||^2).
//
// Fragment layout for 32-bit WMMA operands (ISA 7.12.2):
//   A 16x4 : lane<16 -> {K=k,K=k+1}, lane>=16 -> {K=k+2,K=k+3}, M = lane&15
//   B 4x16 : same K split, N = lane&15
//   C/D    : VGPR v holds M = v + 8*(lane>=16), N = lane&15
//
// Both hot loops reuse each A fragment across TWO accumulators (16x32 tiles)
// to halve A-operand traffic per matrix op (2 WMMA per global_load_b64).

typedef __attribute__((ext_vector_type(2))) float v2f;
typedef __attribute__((ext_vector_type(8))) float v8f;

#define NEG_BIG (-3.402823466e38f)

// --------------------------------------------------------------------------
// Implicit-GEMM conv (k=4, stride=2, pad=1) + bias + ReLU.  NCHW / OIHW.
// Block = 128 threads (4 waves); each wave computes a 16(M) x 32(N) tile
// as two accumulators sharing every A (weight) fragment.
// --------------------------------------------------------------------------
template <int Cin, int H, int W, int Cout, int OH, int OW>
__global__ __launch_bounds__(128) void conv_relu_wmma(
    const float* __restrict__ x, const float* __restrict__ wgt,
    const float* __restrict__ bias, float* __restrict__ out)
{
  constexpr int K     = Cin * 16;
  constexpr int KCH   = (K < 64) ? K : 64;   // 48 for layer 1, else 64
  constexpr int NSTEP = KCH / 4;             // WMMA k-steps per chunk
  constexpr int HWo   = OH * OW;             // pow2 -> shifts
  __shared__ float Bsh[KCH][32];

  const int tid  = threadIdx.x;
  const int lane = tid & 31;
  const int wave = tid >> 5;
  const int lm   = lane & 15;
  const int hi   = lane >> 4;
  const int koff = hi * 2;

  const int n0 = blockIdx.x * 32;                 // spatial tile (32 wide)
  const int m0 = (blockIdx.y * 4 + wave) * 16;    // channel tile (per wave)

  v8f acc0 = {}, acc1 = {};

  for (int k0 = 0; k0 < K; k0 += KCH) {
    // ---- cooperative im2col gather into LDS: Bsh[kk][nn] ----
#pragma unroll
    for (int e = tid; e < KCH * 32; e += 128) {
      const int kk = e >> 5, nn = e & 31;
      const int k  = k0 + kk;
      const int c  = k >> 4, rs = k & 15, r = rs >> 2, s = rs & 3;
      const int n  = n0 + nn;
      const int b  = n / HWo;                 // compile-time pow2 shifts
      const int oh = (n % HWo) / OW;
      const int ow = n % OW;
      const int ih = oh * 2 - 1 + r;
      const int iw = ow * 2 - 1 + s;
      float v = 0.0f;
      if ((unsigned)ih < (unsigned)H && (unsigned)iw < (unsigned)W)
        v = x[((size_t)(b * Cin + c) * H + ih) * W + iw];
      Bsh[kk][nn] = v;
    }
    __syncthreads();

    // ---- batch A loads (NSTEP global_load_b64 in flight) ----
    const float* wrow = &wgt[(size_t)(m0 + lm) * K + k0 + koff];
    v2f afrag[NSTEP];
#pragma unroll
    for (int j = 0; j < NSTEP; ++j)
      afrag[j] = *(const v2f*)&wrow[4 * j];

    // ---- WMMA: each A fragment feeds both N half-tiles ----
#pragma unroll
    for (int j = 0; j < NSTEP; ++j) {
      v2f b0, b1;
      b0.x = Bsh[4 * j + koff][lm];
      b0.y = Bsh[4 * j + koff + 1][lm];
      b1.x = Bsh[4 * j + koff][16 + lm];
      b1.y = Bsh[4 * j + koff + 1][16 + lm];
      acc0 = __builtin_amdgcn_wmma_f32_16x16x4_f32(
          false, afrag[j], false, b0, (short)0, acc0, false, false);
      acc1 = __builtin_amdgcn_wmma_f32_16x16x4_f32(
          false, afrag[j], false, b1, (short)0, acc1, false, false);
    }
    __syncthreads();
  }

  // ---- epilogue: bias + ReLU, scatter to NCHW (compile-time strides) ----
  const int nA = n0 + lm;        // columns 0..15
  const int nB = n0 + 16 + lm;   // columns 16..31
  const int bA = nA / HWo, ohA = (nA % HWo) / OW, owA = nA % OW;
  const int bB = nB / HWo, ohB = (nB % HWo) / OW, owB = nB % OW;
#pragma unroll
  for (int vv = 0; vv < 8; ++vv) {
    const int m = m0 + vv + 8 * hi;
    const float bm = bias[m];
    float r0 = acc0[vv] + bm;  r0 = r0 > 0.0f ? r0 : 0.0f;
    float r1 = acc1[vv] + bm;  r1 = r1 > 0.0f ? r1 : 0.0f;
    out[((size_t)(bA * Cout + m) * OH + ohA) * OW + owA] = r0;
    out[((size_t)(bB * Cout + m) * OH + ohB) * OW + owB] = r1;
  }
}

// --------------------------------------------------------------------------
// halfnorm[k] = 0.5 * ||codebook[k]||^2   (8192 codes x 256 dims)
// --------------------------------------------------------------------------
__global__ __launch_bounds__(256) void codebook_halfnorm(
    const float* __restrict__ cb, float* __restrict__ hn)
{
  const int k = blockIdx.x * blockDim.x + threadIdx.x;   // 8192 threads
  const float* row = cb + (size_t)k * 256;
  float s = 0.0f;
#pragma unroll 4
  for (int c = 0; c < 256; c += 4) {
    const float4 v = *(const float4*)&row[c];
    s += v.x * v.x + v.y * v.y + v.z * v.z + v.w * v.w;
  }
  hn[k] = 0.5f * s;
}

// --------------------------------------------------------------------------
// VQ: one block handles 32 zf rows (two 16-row groups); each wave sweeps
// every 4th tile of 16 codes.  Both row-groups' B fragments are hoisted into
// registers (invariant over code tiles), so the hot loop is codebook loads
// + 2 WMMAs per load.  Codebook L2 traffic: 256 blocks x 8 MB = 2 GB.
// --------------------------------------------------------------------------
__global__ __launch_bounds__(128) void vq_argmin_wmma(
    const float* __restrict__ z4, const float* __restrict__ cb,
    const float* __restrict__ hn, float* __restrict__ zq_out,
    int* __restrict__ idx_out)
{
  __shared__ float Zsh[256][32];     // [dim][row] staged zf tile (32 KB)
  __shared__ float RedV0[128], RedV1[128];
  __shared__ int   RedI0[128], RedI1[128];
  __shared__ int   WinIdx[32];

  const int tid  = threadIdx.x;
  const int lane = tid & 31;
  const int wave = tid >> 5;
  const int lm   = lane & 15;
  const int hi   = lane >> 4;
  const int koff = hi * 2;

  const int n0 = blockIdx.x * 32;    // zf row n = (b, oh, ow), all pow2 dims

  // ---- stage zf tile: Zsh[c][nn] = z4[b][c][oh][ow] ----
#pragma unroll
  for (int e = tid; e < 256 * 32; e += 128) {
    const int c = e >> 5, nn = e & 31;
    const int n = n0 + nn;
    const int b = n >> 8, oh = (n >> 4) & 15, ow = n & 15;
    Zsh[c][nn] = z4[((size_t)(b * 256 + c) * 16 + oh) * 16 + ow];
  }
  __syncthreads();

  // ---- hoist all B fragments for both row-groups into registers ----
  v2f bf0[64], bf1[64];
#pragma unroll
  for (int j = 0; j < 64; ++j) {
    bf0[j].x = Zsh[4 * j + koff][lm];
    bf0[j].y = Zsh[4 * j + koff + 1][lm];
    bf1[j].x = Zsh[4 * j + koff][16 + lm];
    bf1[j].y = Zsh[4 * j + koff + 1][16 + lm];
  }

  float bestv0 = NEG_BIG, bestv1 = NEG_BIG;
  int   besti0 = 0,       besti1 = 0;

  // 512 code tiles of 16; wave-strided so EXEC stays full.
  for (int ct = wave; ct < 512; ct += 4) {
    const int c0 = ct * 16;
    const float* arow = &cb[(size_t)(c0 + lm) * 256 + koff];
    v8f acc0 = {}, acc1 = {};
#pragma unroll
    for (int kb = 0; kb < 4; ++kb) {          // 4 bursts of 16 k-steps
      v2f afrag[16];
#pragma unroll
      for (int j = 0; j < 16; ++j)            // 16 loads in flight
        afrag[j] = *(const v2f*)&arow[kb * 64 + 4 * j];
#pragma unroll
      for (int j = 0; j < 16; ++j) {          // 2 WMMAs per A fragment
        acc0 = __builtin_amdgcn_wmma_f32_16x16x4_f32(
            false, afrag[j], false, bf0[kb * 16 + j], (short)0, acc0,
            false, false);
        acc1 = __builtin_amdgcn_wmma_f32_16x16x4_f32(
            false, afrag[j], false, bf1[kb * 16 + j], (short)0, acc1,
            false, false);
      }
    }
#pragma unroll
    for (int vv = 0; vv < 8; ++vv) {
      const int code = c0 + vv + 8 * hi;
      const float h  = hn[code];
      const float s0 = acc0[vv] - h;           // codes scanned ascending:
      const float s1 = acc1[vv] - h;           // strict > keeps min index
      if (s0 > bestv0) { bestv0 = s0; besti0 = code; }
      if (s1 > bestv1) { bestv1 = s1; besti1 = code; }
    }
  }

  RedV0[tid] = bestv0; RedI0[tid] = besti0;
  RedV1[tid] = bestv1; RedI1[tid] = besti1;
  __syncthreads();

  if (tid < 32) {   // row = tid; combine 4 waves x 2 half-lanes per row
    const int g   = tid >> 4;        // row-group 0 or 1
    const int lm2 = tid & 15;
    const float* RV = g ? RedV1 : RedV0;
    const int*   RI = g ? RedI1 : RedI0;
    float bv = NEG_BIG; int bi = 0;
#pragma unroll
    for (int s = 0; s < 8; ++s) {
      const int src = (s >> 1) * 32 + (s & 1) * 16 + lm2;
      const float v = RV[src]; const int i = RI[src];
      if (v > bv || (v == bv && i < bi)) { bv = v; bi = i; }
    }
    WinIdx[tid] = bi;
    idx_out[n0 + tid] = bi;
  }
  __syncthreads();

  // ---- gather winning codebook rows -> z_q (NCHW), ow-contiguous writes --
#pragma unroll
  for (int e = tid; e < 256 * 32; e += 128) {
    const int c = e >> 5, nn = e & 31;
    const int n = n0 + nn;
    const int b = n >> 8, oh = (n >> 4) & 15, ow = n & 15;
    zq_out[((size_t)(b * 256 + c) * 16 + oh) * 16 + ow] =
        cb[(size_t)WinIdx[nn] * 256 + c];
  }
}

// --------------------------------------------------------------------------
// Host-side launcher.  Workspace ping-pong: r0 = max(z1,z3) = 128 MB,
// r1 = max(z2,z4) = 64 MB, + 32 KB code half-norms (~201 MB, ~L2-resident).
// --------------------------------------------------------------------------
extern "C" void kernel_launch(void* const* d_in, const int* in_sizes, int n_in,
                              void* d_out, int out_size, void* d_ws, size_t ws_size,
                              hipStream_t stream) {
  const float* x  = (const float*)d_in[0];
  const float* w1 = (const float*)d_in[1];
  const float* b1 = (const float*)d_in[2];
  const float* w2 = (const float*)d_in[3];
  const float* b2 = (const float*)d_in[4];
  const float* w3 = (const float*)d_in[5];
  const float* b3 = (const float*)d_in[6];
  const float* w4 = (const float*)d_in[7];
  const float* b4 = (const float*)d_in[8];
  const float* cb = (const float*)d_in[9];

  float* zq  = (float*)d_out;                                      // (32,256,16,16)
  int*   idx = (int*)((float*)d_out + (size_t)32 * 256 * 16 * 16); // (32,16,16)

  char* ws = (char*)d_ws;
  const size_t R0_BYTES = (size_t)32 * 64 * 128 * 128 * 4;  // 128 MB (z1 / z3)
  const size_t R1_BYTES = (size_t)32 * 128 * 64 * 64 * 4;   //  64 MB (z2 / z4)
  float* r0 = (float*)ws;
  float* r1 = (float*)(ws + R0_BYTES);
  float* hn = (float*)(ws + R0_BYTES + R1_BYTES);

  // 0.5 * ||codebook_k||^2 (independent of convs, launch first)
  codebook_halfnorm<<<32, 256, 0, stream>>>(cb, hn);

  // conv1: 3 -> 64,  256^2 -> 128^2, K = 48,  Ntiles(32) = 16384
  conv_relu_wmma<3, 256, 256, 64, 128, 128>
      <<<dim3(16384, 1), 128, 0, stream>>>(x,  w1, b1, r0);
  // conv2: 64 -> 128, 128^2 -> 64^2, K = 1024, Ntiles(32) = 4096
  conv_relu_wmma<64, 128, 128, 128, 64, 64>
      <<<dim3(4096, 2), 128, 0, stream>>>(r0, w2, b2, r1);
  // conv3: 128 -> 256, 64^2 -> 32^2, K = 2048, Ntiles(32) = 1024
  conv_relu_wmma<128, 64, 64, 256, 32, 32>
      <<<dim3(1024, 4), 128, 0, stream>>>(r1, w3, b3, r0);
  // conv4: 256 -> 256, 32^2 -> 16^2, K = 4096, Ntiles(32) = 256
  conv_relu_wmma<256, 32, 32, 256, 16, 16>
      <<<dim3(256, 4), 128, 0, stream>>>(r0, w4, b4, r1);

  // VQ: 8192 zf rows x 8192 codes x 256 dims, 32 rows per block
  vq_argmin_wmma<<<256, 128, 0, stream>>>(r1, cb, hn, zq, idx);
}